// LinkPredictionGraphSAGE_50190987821456
// MI455X (gfx1250) — compile-verified
//
#include <hip/hip_runtime.h>

#define N_NODES 50000
#define N_EDGES 600000
#define IN_CH   128
#define HID_CH  128
#define OUT_CH  64

typedef __attribute__((ext_vector_type(2))) float v2f;
typedef __attribute__((ext_vector_type(8))) float v8f;

// ---------------------------------------------------------------------------
// degree: deg[dst[e]] += 1
// ---------------------------------------------------------------------------
__global__ void sage_deg_kernel(const int* __restrict__ dst,
                                float* __restrict__ deg, int E) {
    int e = blockIdx.x * blockDim.x + threadIdx.x;
    if (e < E) atomicAdd(&deg[dst[e]], 1.0f);
}

// ---------------------------------------------------------------------------
// scatter-add: agg[dst[e]][:] += x[src[e]][:]
// one thread per (edge, 4-channel group); float4 gather + 4 f32 atomics
// ---------------------------------------------------------------------------
__global__ void sage_scatter_kernel(const float* __restrict__ x,
                                    const int* __restrict__ src,
                                    const int* __restrict__ dst,
                                    float* __restrict__ agg,
                                    int E, int C) {
    long long t = (long long)blockIdx.x * blockDim.x + threadIdx.x;
    const int groups = C >> 2;
    if (t >= (long long)E * groups) return;
    int e = (int)(t / groups);
    int g = ((int)(t - (long long)e * groups)) << 2;
    int s = src[e];
    int d = dst[e];
    const float4 v = *reinterpret_cast<const float4*>(x + (long long)s * C + g);
    float* ap = agg + (long long)d * C + g;
    atomicAdd(ap + 0, v.x);
    atomicAdd(ap + 1, v.y);
    atomicAdd(ap + 2, v.z);
    atomicAdd(ap + 3, v.w);
}

// ---------------------------------------------------------------------------
// fused SAGE layer: out[m][n] = act( (agg[m]/max(deg[m],1)) @ Wl
//                                    + xin[m] @ Wr + b )
// grid.x = nNodes/16, block = (32, N/16).  Each wave owns one 16-col tile.
// Uses V_WMMA_F32_16X16X4_F32 (exact f32 path); K-loop fully unrolled
// (K, N compile-time) -> 2*K/4 static WMMAs per wave.
// ---------------------------------------------------------------------------
template <int K, int N, bool DORELU>
__global__ __launch_bounds__(32 * (N / 16))
void sage_layer_kernel(const float* __restrict__ agg,
                       const float* __restrict__ deg,
                       const float* __restrict__ xin,
                       const float* __restrict__ Wl,
                       const float* __restrict__ Wr,
                       const float* __restrict__ bias,
                       float* __restrict__ out,
                       int nNodes) {
    // 16-node tiles of mean and x; pad row stride by 4 floats so that the
    // 16 lanes reading one column hit 16 distinct banks (stride mod 64 == 4).
    __shared__ float sMean[16][K + 4];
    __shared__ float sX[16][K + 4];

    constexpr int NWAVES   = N / 16;
    constexpr int NTHREADS = 32 * NWAVES;

    const int tid   = threadIdx.y * 32 + threadIdx.x;
    const int node0 = blockIdx.x * 16;

    // cooperative stage of A tiles (mean and x) into LDS
    #pragma unroll 2
    for (int i = tid; i < 16 * K; i += NTHREADS) {
        int row = i / K;
        int k   = i - row * K;
        int g   = node0 + row;
        float xv = 0.0f, mv = 0.0f;
        if (g < nNodes) {
            long long base = (long long)g * K;
            xv = xin[base + k];
            float dg = deg[g];
            mv = agg[base + k] * (1.0f / fmaxf(dg, 1.0f));
        }
        sX[row][k]    = xv;
        sMean[row][k] = mv;
    }
    __syncthreads();

    // WMMA fragment coordinates (32-bit A/B layout, ISA 7.12.2):
    //   A 16x4:  lanes 0-15 row M=lane hold K0(v0),K1(v1); lanes 16-31 hold K2,K3
    //   B 4x16:  lanes 0-15 col N=lane hold K0(v0),K1(v1); lanes 16-31 hold K2,K3
    const int lane = threadIdx.x;
    const int row  = lane & 15;
    const int kHi  = lane >> 4;                 // 0 or 1
    const int col  = threadIdx.y * 16 + (lane & 15);

    const float* wl = Wl + col;
    const float* wr = Wr + col;

    v8f acc = {};
    #pragma unroll
    for (int kb = 0; kb < K; kb += 4) {
        const int k0 = kb + 2 * kHi;
        v2f aM, aX, bL, bR;
        aM[0] = sMean[row][k0];
        aM[1] = sMean[row][k0 + 1];
        aX[0] = sX[row][k0];
        aX[1] = sX[row][k0 + 1];
        bL[0] = wl[k0 * N];
        bL[1] = wl[(k0 + 1) * N];
        bR[0] = wr[k0 * N];
        bR[1] = wr[(k0 + 1) * N];
        // D = A*B + C, f32 exact
        acc = __builtin_amdgcn_wmma_f32_16x16x4_f32(false, aM, false, bL,
                                                    (short)0, acc, false, false);
        acc = __builtin_amdgcn_wmma_f32_16x16x4_f32(false, aX, false, bR,
                                                    (short)0, acc, false, false);
    }

    // D layout: VGPR r, lanes 0-15 -> M=r, lanes 16-31 -> M=r+8; N = lane&15
    const float bv = bias[col];
    #pragma unroll
    for (int r = 0; r < 8; ++r) {
        int m = r + 8 * kHi;
        int g = node0 + m;
        if (g < nNodes) {
            float v = acc[r] + bv;
            if (DORELU) v = fmaxf(v, 0.0f);
            out[(long long)g * N + col] = v;
        }
    }
}

// ---------------------------------------------------------------------------
extern "C" void kernel_launch(void* const* d_in, const int* in_sizes, int n_in,
                              void* d_out, int out_size, void* d_ws, size_t ws_size,
                              hipStream_t stream) {
    (void)in_sizes; (void)n_in; (void)out_size; (void)ws_size;

    const float* x   = (const float*)d_in[0];
    const int*   ei  = (const int*)d_in[1];
    const float* W1l = (const float*)d_in[2];
    const float* W1r = (const float*)d_in[3];
    const float* b1  = (const float*)d_in[4];
    const float* W2l = (const float*)d_in[5];
    const float* W2r = (const float*)d_in[6];
    const float* b2  = (const float*)d_in[7];
    float*       out = (float*)d_out;

    const int N = N_NODES;
    const int E = N_EDGES;
    const int* src = ei;         // edge_index[0]
    const int* dst = ei + E;     // edge_index[1]

    // workspace: agg [N*128] | h [N*128] | deg [N]
    float* agg = (float*)d_ws;
    float* h   = agg + (size_t)N * HID_CH;
    float* deg = h   + (size_t)N * HID_CH;

    hipMemsetAsync(agg, 0, (size_t)N * IN_CH * sizeof(float), stream);
    hipMemsetAsync(deg, 0, (size_t)N * sizeof(float), stream);

    sage_deg_kernel<<<(E + 255) / 256, 256, 0, stream>>>(dst, deg, E);

    // ----- layer 1 -----
    {
        long long tot = (long long)E * (IN_CH / 4);
        int blocks = (int)((tot + 255) / 256);
        sage_scatter_kernel<<<blocks, 256, 0, stream>>>(x, src, dst, agg, E, IN_CH);
        dim3 blk(32, HID_CH / 16);   // 8 waves
        sage_layer_kernel<IN_CH, HID_CH, true><<<(N + 15) / 16, blk, 0, stream>>>(
            agg, deg, x, W1l, W1r, b1, h, N);
    }

    // ----- layer 2 -----
    {
        hipMemsetAsync(agg, 0, (size_t)N * HID_CH * sizeof(float), stream);
        long long tot = (long long)E * (HID_CH / 4);
        int blocks = (int)((tot + 255) / 256);
        sage_scatter_kernel<<<blocks, 256, 0, stream>>>(h, src, dst, agg, E, HID_CH);
        dim3 blk(32, OUT_CH / 16);   // 4 waves
        sage_layer_kernel<HID_CH, OUT_CH, false><<<(N + 15) / 16, blk, 0, stream>>>(
            agg, deg, h, W2l, W2r, b2, out, N);
    }
}